// Attention_17738214932808
// MI455X (gfx1250) — compile-verified
//
#include <hip/hip_runtime.h>
#include <hip/hip_bf16.h>
#include <math.h>

// ---------------------------------------------------------------------------
// Problem constants (reference: B=64, T=4096, D=256, fp32 everywhere)
// ---------------------------------------------------------------------------
#define B_SZ 64
#define T_SZ 4096
#define D_SZ 256

typedef float v2f __attribute__((ext_vector_type(2)));
typedef float v8f __attribute__((ext_vector_type(8)));

// ---------------------------------------------------------------------------
// Kernel A: mids[b,d] = sum_e W[d,e] * q[b,e]   (exact-fp32 WMMA GEMM)
//   D = A x B + C with M=batch(64), N=d(256), K=e(256)
//   A[m,k] = q[mBase+m, k],  B[k,n] = W[nBase+n, k]
//   grid (B/16, D/16) = (4,16), block = 32 (one wave per 16x16 output tile)
// ---------------------------------------------------------------------------
__global__ __launch_bounds__(32)
void mids_wmma_kernel(const float* __restrict__ q,
                      const float* __restrict__ W,
                      float* __restrict__ mids)
{
    const int mBase = blockIdx.x * 16;   // batch rows
    const int nBase = blockIdx.y * 16;   // output-feature rows of W
    const int lane  = threadIdx.x;       // 0..31 (wave32)
    const int half  = lane >> 4;         // 0: K pair {0,1}, 1: K pair {2,3}
    const int lm    = lane & 15;

    const float* qrow = q + (size_t)(mBase + lm) * D_SZ + 2 * half;
    const float* wrow = W + (size_t)(nBase + lm) * D_SZ + 2 * half;

    v8f c = {};
    #pragma unroll 8
    for (int k = 0; k < D_SZ; k += 4) {
        // A 16x4 fp32: VGPR0 = K0(lanes0-15)/K2(lanes16-31); VGPR1 = K1/K3
        v2f a; a.x = qrow[k];  a.y = qrow[k + 1];
        // B 4x16 fp32: VGPR0 = row K0 (lanes0-15) / row K2 (lanes16-31); VGPR1 = K1/K3
        v2f b; b.x = wrow[k];  b.y = wrow[k + 1];
        c = __builtin_amdgcn_wmma_f32_16x16x4_f32(
                /*neg_a=*/false, a, /*neg_b=*/false, b,
                /*c_mod=*/(short)0, c, /*reuse_a=*/false, /*reuse_b=*/false);
    }

    // C/D layout: VGPR r -> M = r + 8*half, N = lm
    #pragma unroll
    for (int r = 0; r < 8; ++r) {
        mids[(size_t)(mBase + 8 * half + r) * D_SZ + nBase + lm] = c[r];
    }
}

// ---------------------------------------------------------------------------
// Kernel B: scores[b,t] = tanh( K[b,t,:] . mids[b,:] + bias )
//   Pure bandwidth streamer over k (268 MB). One row per wave; each lane
//   loads 8 contiguous floats (2 x float4, fully coalesced 1KB/row/wave),
//   FMAs against a cached slice of mids, then a 5-step shfl_xor reduction.
//   grid (T/8, B), block = 256 (8 waves -> 8 rows per block)
// ---------------------------------------------------------------------------
__global__ __launch_bounds__(256)
void scores_kernel(const float* __restrict__ kmat,
                   const float* __restrict__ mids,
                   const float* __restrict__ bias,
                   float* __restrict__ scores)
{
    __shared__ float s_mids[D_SZ];
    const int b   = blockIdx.y;
    const int tid = threadIdx.x;         // 0..255
    s_mids[tid] = mids[(size_t)b * D_SZ + tid];
    __syncthreads();

    const int lane = tid & 31;
    const int wave = tid >> 5;
    const int t    = blockIdx.x * 8 + wave;

    const float4* row =
        (const float4*)(kmat + ((size_t)b * T_SZ + t) * D_SZ + lane * 8);
    float4 r0 = row[0];
    float4 r1 = row[1];

    const float* md = &s_mids[lane * 8];
    float acc = r0.x * md[0] + r0.y * md[1] + r0.z * md[2] + r0.w * md[3]
              + r1.x * md[4] + r1.y * md[5] + r1.z * md[6] + r1.w * md[7];

    #pragma unroll
    for (int off = 16; off > 0; off >>= 1)
        acc += __shfl_xor(acc, off, 32);

    if (lane == 0)
        scores[(size_t)b * T_SZ + t] = tanhf(acc + bias[0]);
}

// ---------------------------------------------------------------------------
// Kernel C: masked, max-stabilized softmax over T per batch.
//   grid = B, block = 1024 (32 waves). Scores staged in 16KB LDS.
// ---------------------------------------------------------------------------
__global__ __launch_bounds__(1024)
void softmax_kernel(const float* __restrict__ scores,
                    const float* __restrict__ mask,
                    float* __restrict__ out)
{
    __shared__ float s_sc[T_SZ];
    __shared__ float s_red[32];

    const int b    = blockIdx.x;
    const int tid  = threadIdx.x;        // 0..1023
    const int lane = tid & 31;
    const int wave = tid >> 5;

    const float* sb = scores + (size_t)b * T_SZ;
    const float* mb = mask   + (size_t)b * T_SZ;
    float*       ob = out    + (size_t)b * T_SZ;

    // ---- load scores into LDS, tracking partial max ----
    float mx = -INFINITY;
    #pragma unroll
    for (int t = tid; t < T_SZ; t += 1024) {
        float v = sb[t];
        s_sc[t] = v;
        mx = fmaxf(mx, v);
    }
    #pragma unroll
    for (int off = 16; off > 0; off >>= 1)
        mx = fmaxf(mx, __shfl_xor(mx, off, 32));
    if (lane == 0) s_red[wave] = mx;
    __syncthreads();
    if (tid < 32) {
        float v = s_red[tid];
        #pragma unroll
        for (int off = 16; off > 0; off >>= 1)
            v = fmaxf(v, __shfl_xor(v, off, 32));
        if (tid == 0) s_red[0] = v;
    }
    __syncthreads();
    mx = s_red[0];
    __syncthreads();   // everyone has read s_red[0]; safe to reuse below

    // ---- e = exp(s - max) * m; accumulate sum; stash e back in LDS ----
    float sum = 0.0f;
    #pragma unroll
    for (int t = tid; t < T_SZ; t += 1024) {
        float e = __expf(s_sc[t] - mx) * mb[t];
        s_sc[t] = e;
        sum += e;
    }
    #pragma unroll
    for (int off = 16; off > 0; off >>= 1)
        sum += __shfl_xor(sum, off, 32);
    if (lane == 0) s_red[wave] = sum;
    __syncthreads();
    if (tid < 32) {
        float v = s_red[tid];
        #pragma unroll
        for (int off = 16; off > 0; off >>= 1)
            v += __shfl_xor(v, off, 32);
        if (tid == 0) s_red[0] = v;
    }
    __syncthreads();
    const float inv = 1.0f / s_red[0];

    #pragma unroll
    for (int t = tid; t < T_SZ; t += 1024)
        ob[t] = s_sc[t] * inv;
}

// ---------------------------------------------------------------------------
// Launch
//   d_in order: q[B,D], k[B,T,D], m[B,T], W[D,D], bias[1]   (all fp32)
//   d_out: attn[B,T] fp32
//   d_ws layout: mids (B*D floats) | scores (B*T floats)  ~= 1.06 MB
// ---------------------------------------------------------------------------
extern "C" void kernel_launch(void* const* d_in, const int* in_sizes, int n_in,
                              void* d_out, int out_size, void* d_ws, size_t ws_size,
                              hipStream_t stream)
{
    const float* q    = (const float*)d_in[0];
    const float* kmat = (const float*)d_in[1];
    const float* m    = (const float*)d_in[2];
    const float* W    = (const float*)d_in[3];
    const float* bias = (const float*)d_in[4];
    float*       out  = (float*)d_out;

    float* mids   = (float*)d_ws;                    // B*D = 16384 floats
    float* scores = (float*)d_ws + (B_SZ * D_SZ);    // B*T = 262144 floats

    // A: fp32 WMMA GEMM for mids
    mids_wmma_kernel<<<dim3(B_SZ / 16, D_SZ / 16), 32, 0, stream>>>(q, W, mids);

    // B: bandwidth-bound batched GEMV + tanh
    scores_kernel<<<dim3(T_SZ / 8, B_SZ), 256, 0, stream>>>(kmat, mids, bias, scores);

    // C: masked softmax per batch
    softmax_kernel<<<dim3(B_SZ), 1024, 0, stream>>>(scores, m, out);
}